// SphereInterLoss_32177894981699
// MI455X (gfx1250) — compile-verified
//
#include <hip/hip_runtime.h>
#include <math.h>

typedef float v2f __attribute__((ext_vector_type(2)));
typedef float v8f __attribute__((ext_vector_type(8)));

#define NSPH   4096
#define NBATCH 8
#define KSEL   11            // K+1 smallest kept (rank 0 = self, dropped)
#define TILES  (NSPH / 16)
#define TSTRIDE 20           // tile row stride: 16B-aligned rows, conflict-free writes
#define FLTMAX 3.402823466e38f

// Wave-local LDS sync: CDNA5 executes a wave's DS ops in order, so same-wave
// cross-lane store->load visibility needs only a compiler fence + dscnt wait
// (no workgroup s_barrier needed; each wave owns its scratch region).
__device__ __forceinline__ void wave_lds_sync() {
    asm volatile("s_wait_dscnt 0" ::: "memory");
}

// One wave (32 lanes) handles 16 rows of one batch, scanning all 4096 columns
// in 16-wide tiles via V_WMMA_F32_16X16X4_F32 (d2 = C + (-2A)*B).
__global__ __launch_bounds__(256) void sphere_topk_kernel(
    const float4* __restrict__ sph,     // [8][4096] (x,y,z,r)
    float* __restrict__ topdist)        // [8][4096]
{
    // per-wave scratch: 16x16 d2 tile (stride 20) reused later as 16x44 merge area
    __shared__ __align__(16) float scratch[8][704];

    const int tid  = threadIdx.x;
    const int lane = tid & 31;          // wave32 on gfx1250
    const int wave = tid >> 5;
    const int hi   = lane >> 4;         // 0: lanes 0-15, 1: lanes 16-31
    const int lo   = lane & 15;

    const int batch   = blockIdx.x >> 5;            // 8 batches
    const int rowBlk  = blockIdx.x & 31;            // 32 row-blocks of 128 rows
    const int rowBase = rowBlk * 128 + wave * 16;   // this wave's 16 rows
    const long bOff   = (long)batch * NSPH;

    // ---- A fragment: A[m][k] = -2*c[m][k], k in {x,y,z,0} (16x4 f32, 2 VGPR) ----
    float4 si = sph[bOff + rowBase + lo];
    v2f afrag;
    afrag[0] = -2.0f * (hi ? si.z : si.x);
    afrag[1] = hi ? 0.0f : (-2.0f * si.y);

    // ---- srow[r] = |c|^2 of row (hi*8 + r) of this wave's tile ----
    v8f srow;
    #pragma unroll
    for (int r = 0; r < 8; ++r) {
        float4 s = sph[bOff + rowBase + hi * 8 + r];
        srow[r] = s.x * s.x + s.y * s.y + s.z * s.z;
    }

    // ---- per-(row, column-half) running top-11 (sorted ascending) ----
    float keys[KSEL];
    int   idxs[KSEL];
    #pragma unroll
    for (int t = 0; t < KSEL; ++t) { keys[t] = FLTMAX; idxs[t] = 0; }

    float* ts = &scratch[wave][0];
    const int rdOff = lo * TSTRIDE + hi * 8;   // 16B-aligned row segment

    for (int tile = 0; tile < TILES; ++tile) {
        int j = tile * 16 + lo;
        float4 sj = sph[bOff + j];
        __builtin_prefetch(&sph[bOff + ((tile + 1) & (TILES - 1)) * 16 + lo], 0, 3);

        float sqj = sj.x * sj.x + sj.y * sj.y + sj.z * sj.z;

        // B fragment: B[k][n] = c_j[n][k] (4x16 f32, 2 VGPR)
        v2f bfrag;
        bfrag[0] = hi ? sj.z : sj.x;
        bfrag[1] = hi ? 0.0f : sj.y;

        // C fragment: C[m][n] = sq_i[m] + sq_j[n]
        v8f cfrag;
        #pragma unroll
        for (int r = 0; r < 8; ++r) cfrag[r] = srow[r] + sqj;

        // d2 tile = C + (-2A)*B  -> v_wmma_f32_16x16x4_f32
        v8f d = __builtin_amdgcn_wmma_f32_16x16x4_f32(
            false, afrag, false, bfrag, (short)0, cfrag, false, false);

        // transpose through this wave's LDS pad: ts[m*20 + n] = d2(row m, col n)
        #pragma unroll
        for (int r = 0; r < 8; ++r)
            ts[(r + 8 * hi) * TSTRIDE + lo] = d[r];
        wave_lds_sync();

        // selection: lane handles row `lo`, columns (hi*8 .. hi*8+7) of this tile
        const float4* rowp = reinterpret_cast<const float4*>(&ts[rdOff]);
        float4 v0 = rowp[0];                 // ds_load_b128
        float4 v1 = rowp[1];                 // ds_load_b128
        float cand[8] = { v0.x, v0.y, v0.z, v0.w, v1.x, v1.y, v1.z, v1.w };

        int cbase = tile * 16 + hi * 8;
        #pragma unroll
        for (int q = 0; q < 8; ++q) {
            float key = cand[q];
            if (key < keys[KSEL - 1]) {
                keys[KSEL - 1] = key;
                idxs[KSEL - 1] = cbase + q;
                #pragma unroll
                for (int t = KSEL - 1; t >= 1; --t) {
                    bool sw = keys[t] < keys[t - 1];
                    float fk = keys[t - 1]; int fi2 = idxs[t - 1];
                    keys[t - 1] = sw ? keys[t] : keys[t - 1];
                    idxs[t - 1] = sw ? idxs[t] : idxs[t - 1];
                    keys[t]     = sw ? fk      : keys[t];
                    idxs[t]     = sw ? fi2     : idxs[t];
                }
            }
        }
        wave_lds_sync();   // order this tile's reads before next tile's writes
    }

    // ---- merge the two column-half lists per row via LDS (wave-local) ----
    {
        int kb = lo * 44 + hi * 11;
        int ib = lo * 44 + 22 + hi * 11;
        #pragma unroll
        for (int t = 0; t < KSEL; ++t) {
            ts[kb + t] = keys[t];
            ts[ib + t] = __int_as_float(idxs[t]);
        }
    }
    wave_lds_sync();

    if (hi == 0) {
        float fk[KSEL]; int fi[KSEL];
        #pragma unroll
        for (int t = 0; t < KSEL; ++t) { fk[t] = FLTMAX; fi[t] = 0; }
        for (int q = 0; q < 2 * KSEL; ++q) {
            float key = ts[lo * 44 + q];
            int   id  = __float_as_int(ts[lo * 44 + 22 + q]);
            if (key < fk[KSEL - 1]) {
                fk[KSEL - 1] = key;
                fi[KSEL - 1] = id;
                #pragma unroll
                for (int t = KSEL - 1; t >= 1; --t) {
                    bool sw = fk[t] < fk[t - 1];
                    float a = fk[t - 1]; int b = fi[t - 1];
                    fk[t - 1] = sw ? fk[t] : fk[t - 1];
                    fi[t - 1] = sw ? fi[t] : fi[t - 1];
                    fk[t]     = sw ? a     : fk[t];
                    fi[t]     = sw ? b     : fi[t];
                }
            }
        }
        // drop rank 0 (self); exact distance recompute for ranks 1..10
        float4 ci = sph[bOff + rowBase + lo];
        float ri = ci.w;
        float best = FLTMAX;
        #pragma unroll
        for (int t = 1; t < KSEL; ++t) {
            float4 cj = sph[bOff + fi[t]];
            float dx = ci.x - cj.x, dy = ci.y - cj.y, dz = ci.z - cj.z;
            float dis = sqrtf(dx * dx + dy * dy + dz * dz);
            best = fminf(best, dis - ri - cj.w);
        }
        topdist[bOff + rowBase + lo] = best;
    }
}

// Single-block two-pass variance reduction: var(ddof=1) per batch, mean over batches.
__global__ __launch_bounds__(256) void finalize_kernel(
    const float* __restrict__ topdist, float* __restrict__ out)
{
    __shared__ float red[256];
    __shared__ float bc;
    const int tid = threadIdx.x;
    float varSum = 0.0f;

    for (int b = 0; b < NBATCH; ++b) {
        const float* x = topdist + b * NSPH;

        float s = 0.0f;
        for (int i = tid; i < NSPH; i += 256) s += x[i];
        red[tid] = s;
        __syncthreads();
        for (int off = 128; off > 0; off >>= 1) {
            if (tid < off) red[tid] += red[tid + off];
            __syncthreads();
        }
        if (tid == 0) bc = red[0] * (1.0f / NSPH);
        __syncthreads();
        float mean = bc;

        float s2 = 0.0f;
        for (int i = tid; i < NSPH; i += 256) { float d = x[i] - mean; s2 += d * d; }
        red[tid] = s2;
        __syncthreads();
        for (int off = 128; off > 0; off >>= 1) {
            if (tid < off) red[tid] += red[tid + off];
            __syncthreads();
        }
        if (tid == 0) varSum += red[0] * (1.0f / (NSPH - 1));
        __syncthreads();
    }
    if (tid == 0) out[0] = varSum * (1.0f / NBATCH);
}

extern "C" void kernel_launch(void* const* d_in, const int* in_sizes, int n_in,
                              void* d_out, int out_size, void* d_ws, size_t ws_size,
                              hipStream_t stream) {
    const float4* sph = reinterpret_cast<const float4*>(d_in[0]); // [8][4096][4] f32
    float* topdist = reinterpret_cast<float*>(d_ws);              // 8*4096 floats
    float* out     = reinterpret_cast<float*>(d_out);             // scalar

    sphere_topk_kernel<<<NBATCH * 32, 256, 0, stream>>>(sph, topdist);
    finalize_kernel<<<1, 256, 0, stream>>>(topdist, out);
}